// ShapeMatchingLoss_59356448030933
// MI455X (gfx1250) — compile-verified
//
#include <hip/hip_runtime.h>

// ---------------------------------------------------------------------------
// ShapeMatchingLoss on gfx1250 (MI455X).
// Latency-bound micro-workload: 2 workgroups, LDS-staged, f32 throughout.
// CDNA5 paths used:
//   * V_WMMA_F32_16X16X4_F32 for the 2x2 rotation applied to 2xN points
//   * global_prefetch_b8 (via __builtin_prefetch) to pull the ~2MB gather
//     footprint toward the caches before the dependent gathers issue.
// ---------------------------------------------------------------------------

#define WCONT 1048576
#define MAXN  1400            // max contour length per range (ranges ~1.0 / 0.001)
#define PI_F   3.1416f
#define STEP_F 0.001f
#define NTHREADS 256

typedef __attribute__((ext_vector_type(2))) float v2f;
typedef __attribute__((ext_vector_type(8))) float v8f;

struct Comm {
  int   na, nb, w, mode;      // mode: 0 = equal, 1 = resample B -> na, 2 = resample TA -> nb
  float d0, d1, sc, ss;
};

__device__ __forceinline__ int iclamp(int v, int lo, int hi) {
  return v < lo ? lo : (v > hi ? hi : v);
}

// lin_sample: pos = (c+1)*0.5*(W-1); zero outside [0, W-1]
__device__ __forceinline__ float lin_sample(const float* __restrict__ row, float c) {
  float pos  = (c + 1.0f) * 0.5f * (float)(WCONT - 1);
  float fl   = floorf(pos);
  float frac = pos - fl;
  // convert via clamped float to avoid UB on wild coords
  int i0 = (int)fminf(fmaxf(fl, -2.0f), (float)WCONT + 2.0f);
  int ia = iclamp(i0,     0, WCONT - 1);
  int ib = iclamp(i0 + 1, 0, WCONT - 1);
  float g0 = (i0 >= 0     && i0     < WCONT) ? row[ia] : 0.0f;
  float g1 = (i0 + 1 >= 0 && i0 + 1 < WCONT) ? row[ib] : 0.0f;
  return g0 * (1.0f - frac) + g1 * frac;
}

// _cue on a 2xN array stored as arr[0..N-1] (row0), arr[MAXN..] (row1)
__device__ __forceinline__ void cue2(const float* __restrict__ arr,
                                     int im, int ii, int ip,
                                     float* out, bool* valid) {
  float v0x = arr[im], v0y = arr[MAXN + im];
  float v1x = arr[ii], v1y = arr[MAXN + ii];
  float v2x = arr[ip], v2y = arr[MAXN + ip];
  bool eq01 = (v0x == v1x) && (v0y == v1y);
  bool eq12 = (v1x == v2x) && (v1y == v2y);
  bool eq02 = (v0x == v2x) && (v0y == v2y);
  bool val  = !(eq01 || eq12 || eq02);
  float ax = v1x - v0x, ay = v1y - v0y;
  float bx = v2x - v1x, by = v2y - v1y;
  float denom = val ? (sqrtf(ax * ax + ay * ay) * sqrtf(bx * bx + by * by)) : 1.0f;
  float cs    = (ax * bx + ay * by) / denom;
  float cross = ax * by - ay * bx;
  float cc    = fminf(fmaxf(cs, -1.0f + 1e-7f), 1.0f - 1e-7f);
  float theta = acosf(cc);
  float sg    = (cross > 0.0f) ? 1.0f : ((cross < 0.0f) ? -1.0f : 0.0f);
  *out   = sg * (PI_F - theta);
  *valid = val;
}

__global__ __launch_bounds__(NTHREADS)
void shape_loss_kernel(const float* __restrict__ pr_a,
                       const float* __restrict__ pr_b,
                       const float* __restrict__ in_a,
                       const float* __restrict__ in_b,
                       float* __restrict__ out,
                       float* __restrict__ ws) {
  __shared__ float sh_coords[MAXN];
  __shared__ float sh_A [2 * MAXN];
  __shared__ float sh_B [2 * MAXN];
  __shared__ float sh_TA[2 * MAXN];
  __shared__ float sh_RS[2 * MAXN];
  __shared__ Comm  cm;

  const int tid = threadIdx.x;
  const float* pr = (blockIdx.x == 0) ? pr_a : pr_b;

  // ---- per-thread scalar params (all threads read the 4 floats) ----
  float p0 = pr[0], p1 = pr[1], p2 = pr[2], p3 = pr[3];
  float a0 = fminf(p0, p1), a1 = fmaxf(p0, p1);
  float b0 = fminf(p2, p3), b1 = fmaxf(p2, p3);

  const float* rowA0 = in_a;
  const float* rowA1 = in_a + WCONT;
  const float* rowB0 = in_b;
  const float* rowB1 = in_b + WCONT;

  // ---- prefetch the gather footprint (global_prefetch_b8) ----
  {
    float lims[4][2] = {{a0, a1}, {a0, a1}, {b0, b1}, {b0, b1}};
    const float* rows[4] = {rowA0, rowA1, rowB0, rowB1};
#pragma unroll
    for (int r = 0; r < 4; ++r) {
      float pl = (fminf(lims[r][0], lims[r][1]) + 1.0f) * 0.5f * (float)(WCONT - 1);
      float ph = (fmaxf(lims[r][0], lims[r][1]) + 1.0f) * 0.5f * (float)(WCONT - 1);
      long lo = (long)fmaxf(0.0f, floorf(pl));
      long hi = (long)fminf((float)(WCONT - 1), ceilf(ph) + 1.0f);
      for (long p = lo + (long)tid * 32; p <= hi; p += (long)NTHREADS * 32)
        __builtin_prefetch(rows[r] + p, 0, 3);
    }
  }

  // ---- build coords for A-range (serial f32 accumulation, must match numpy) ----
  if (tid == 0) {
    int n = 0;
    float t = a0;
    while (t < a1 && n < MAXN - 2) {
      sh_coords[1 + n] = a0 + (t - a0);   // c0 + offs[n]
      n++;
      t = t + STEP_F;                     // f32 round-to-nearest-even
    }
    sh_coords[0]     = a0;
    sh_coords[1 + n] = a1;
    cm.na = n + 2;
  }
  __syncthreads();

  // ---- sample A = partial_contour(input_a, a-range) ----
  {
    int na = cm.na;
    for (int i = tid; i < na; i += NTHREADS) {
      float c = sh_coords[i];
      sh_A[i]        = lin_sample(rowA0, c);
      sh_A[MAXN + i] = lin_sample(rowA1, c);
    }
  }
  __syncthreads();

  // ---- build coords for B-range ----
  if (tid == 0) {
    int n = 0;
    float t = b0;
    while (t < b1 && n < MAXN - 2) {
      sh_coords[1 + n] = b0 + (t - b0);
      n++;
      t = t + STEP_F;
    }
    sh_coords[0]     = b0;
    sh_coords[1 + n] = b1;
    cm.nb = n + 2;
  }
  __syncthreads();

  // ---- sample B = partial_contour(input_b, b-range) ----
  {
    int nb = cm.nb;
    for (int i = tid; i < nb; i += NTHREADS) {
      float c = sh_coords[i];
      sh_B[i]        = lin_sample(rowB0, c);
      sh_B[MAXN + i] = lin_sample(rowB1, c);
    }
  }
  __syncthreads();

  // ---- similarity transform scalars (thread 0) ----
  if (tid == 0) {
    int na = cm.na, nb = cm.nb;
    float A0f = sh_A[0],        A1f = sh_A[MAXN];
    float A0l = sh_A[na - 1],   A1l = sh_A[MAXN + na - 1];
    float B0f = sh_B[0],        B1f = sh_B[MAXN];
    float B0l = sh_B[nb - 1],   B1l = sh_B[MAXN + nb - 1];
    float d0 = B0f - A0f, d1 = B1f - A1f;
    float e00 = (A0l + d0) - B0f, e01 = (A1l + d1) - B1f;   // == A_last - A_first
    float e10 = B0l - B0f,        e11 = B1l - B1f;
    float n0 = sqrtf(e00 * e00 + e01 * e01);
    float n1 = sqrtf(e10 * e10 + e11 * e11);
    float cosv = (e00 * e10 + e01 * e11) / (n0 * n1);
    float sinv = sqrtf(1.0f - cosv * cosv);
    if (!(sinv == sinv)) sinv = 0.0f;                       // NaN -> 0
    float cross = e00 * e11 - e01 * e10;
    float dA0 = A0l - A0f, dA1 = A1l - A1f;
    float scale = n1 / sqrtf(dA0 * dA0 + dA1 * dA1);
    float sgn = (cross < 0.0f) ? -1.0f : 1.0f;
    cm.sc = scale * cosv;
    cm.ss = sgn * scale * sinv;
    cm.d0 = d0;
    cm.d1 = d1;
    // resample mode
    if (na > nb)      { cm.mode = 1; cm.w = na; }
    else if (na < nb) { cm.mode = 2; cm.w = nb; }
    else              { cm.mode = 0; cm.w = na; }
  }
  __syncthreads();

  // ---- TA = R @ A + d via V_WMMA_F32_16X16X4_F32 ----
  // A-operand (16x4): row0 = [sc,-ss,0,0], row1 = [ss,sc,0,0], rest 0.
  //   lanes 0-15 hold K0/K1 (VGPR0/1), lanes 16-31 hold K2/K3 (zero here).
  // B-operand (4x16): per 16-column tile, lane n<16: K0 = A0[col], K1 = A1[col].
  // D rows 0/1 (acc[0], acc[1] on lanes 0-15) are the rotated points.
  {
    const int wave = tid >> 5, lane = tid & 31;
    const int nwaves = NTHREADS / 32;
    float sc = cm.sc, ss = cm.ss, dd0 = cm.d0, dd1 = cm.d1;
    int na = cm.na;
    v2f amat;
    amat[0] = (lane == 0) ? sc  : ((lane == 1) ? ss : 0.0f);
    amat[1] = (lane == 0) ? -ss : ((lane == 1) ? sc : 0.0f);
    for (int col0 = wave * 16; col0 < na; col0 += nwaves * 16) {   // wave-uniform
      int col = col0 + (lane & 15);
      int idx = col < na ? col : (na - 1);
      float msk = ((lane < 16) && (col < na)) ? 1.0f : 0.0f;
      v2f bmat;
      bmat[0] = sh_A[idx] * msk;
      bmat[1] = sh_A[MAXN + idx] * msk;
      v8f acc = {};
      acc = __builtin_amdgcn_wmma_f32_16x16x4_f32(
          /*neg_a=*/false, amat, /*neg_b=*/false, bmat,
          /*c_mod=*/(short)0, acc, /*reuse_a=*/false, /*reuse_b=*/false);
      if ((lane < 16) && (col < na)) {
        sh_TA[col]        = acc[0] + dd0;
        sh_TA[MAXN + col] = acc[1] + dd1;
      }
    }
  }
  __syncthreads();

  // ---- resample the shorter of {B, TA} to width w ----
  {
    int mode = cm.mode;
    if (mode != 0) {
      const float* src = (mode == 1) ? sh_B : sh_TA;
      int w_in  = (mode == 1) ? cm.nb : cm.na;
      int w_out = cm.w;
      float ratio = (float)(w_in - 1) / (float)(w_out - 1);
      for (int j = tid; j < w_out; j += NTHREADS) {
        float pos = (float)j * ratio;
        int i0 = (int)floorf(pos);
        i0 = iclamp(i0, 0, w_in - 2);
        float frac = pos - (float)i0;
        sh_RS[j]        = src[i0] * (1.0f - frac) + src[i0 + 1] * frac;
        sh_RS[MAXN + j] = src[MAXN + i0] * (1.0f - frac) + src[MAXN + i0 + 1] * frac;
      }
    }
  }
  __syncthreads();

  // ---- curvature-cue loss (thread 0) ----
  if (tid == 0) {
    int w = cm.w, mode = cm.mode;
    const float* cueA = (mode == 2) ? sh_RS : sh_TA;
    const float* cueB = (mode == 1) ? sh_RS : sh_B;
    float loss = 1.0f;
    if (w >= 3) {
      float fab = 0.0f;
      for (int i = 1; i <= 9; ++i) {
        int im = iclamp(i - 1, 0, w - 1);
        int ii = iclamp(i,     0, w - 1);
        int ip = iclamp(i + 1, 0, w - 1);
        float ca, cb; bool va, vb;
        cue2(cueA, im, ii, ip, &ca, &va);
        cue2(cueB, im, ii, ip, &cb, &vb);
        bool mask = va && vb && (ca != 0.0f) && (cb != 0.0f);
        float prod = ca * cb;
        float sp = (prod > 0.0f) ? 1.0f : ((prod < 0.0f) ? -1.0f : 0.0f);
        fab += mask ? sp * sqrtf(fabsf(prod)) : 0.0f;
      }
      loss = loss + fab / (float)w;
    }
    ws[blockIdx.x] = loss;
  }
  __syncthreads();

  // ---- block 1 (pr_b) writes sa = A and sb = B2 ----
  if (blockIdx.x == 1) {
    int na = cm.na, w = cm.w, mode = cm.mode;
    const float* b2 = (mode == 1) ? sh_RS : sh_B;
    for (int i = tid; i < na; i += NTHREADS) {
      out[1 + i]      = sh_A[i];
      out[1 + na + i] = sh_A[MAXN + i];
    }
    int base = 1 + 2 * na;
    for (int j = tid; j < w; j += NTHREADS) {
      out[base + j]     = b2[j];
      out[base + w + j] = b2[MAXN + j];
    }
  }
}

__global__ void combine_kernel(const float* __restrict__ ws, float* __restrict__ out) {
  float la = ws[0], lb = ws[1];
  out[0] = (la + lb) * 0.5f + (la - lb) * (la - lb);
}

extern "C" void kernel_launch(void* const* d_in, const int* in_sizes, int n_in,
                              void* d_out, int out_size, void* d_ws, size_t ws_size,
                              hipStream_t stream) {
  const float* pr_a = (const float*)d_in[0];
  const float* pr_b = (const float*)d_in[1];
  const float* in_a = (const float*)d_in[2];
  const float* in_b = (const float*)d_in[3];
  float* out = (float*)d_out;
  float* ws  = (float*)d_ws;
  (void)in_sizes; (void)n_in; (void)out_size; (void)ws_size;

  shape_loss_kernel<<<2, NTHREADS, 0, stream>>>(pr_a, pr_b, in_a, in_b, out, ws);
  combine_kernel<<<1, 1, 0, stream>>>(ws, out);
}